// ASD_26491358282344
// MI455X (gfx1250) — compile-verified
//
#include <hip/hip_runtime.h>
#include <math.h>

// ---------------------------------------------------------------------------
// ASSD between two 16384-point clouds via V_WMMA_F32_16X16X4_F32.
//
// d2[q][r] = |q|^2 + |r|^2 - 2 q.r
// A row   = [-2qx, -2qy, -2qz, 0]        (query, "Q-form")
// B col   = [ rx,   ry,   rz, |r|^2]     (reference, "R-form")
// WMMA -> t = -2 q.r (A's 0 zeroes |r|^2) ;  value = t + |q|^2 (|q|^2 const per row,
// so we min raw t across all column tiles and add |q|^2 once at the end).
// ---------------------------------------------------------------------------

typedef __attribute__((ext_vector_type(2))) float v2f;
typedef __attribute__((ext_vector_type(8))) float v8f;

#define NPTS            16384
#define NTILES          (NPTS / 16)         // 1024 column tiles per sweep
#define TOTAL_WAVES     (2 * NTILES)        // 1024 row-strips x 2 directions
#define WAVES_PER_BLOCK 8
#define BLOCK_THREADS   (WAVES_PER_BLOCK * 32)

// ---- pack both clouds into Q-form and R-form float4 arrays -----------------
__global__ void asd_prep(const float* __restrict__ pred,
                         const float* __restrict__ real,
                         float4* __restrict__ predQ, float4* __restrict__ predR,
                         float4* __restrict__ realQ, float4* __restrict__ realR)
{
    int i = blockIdx.x * blockDim.x + threadIdx.x;
    if (i < NPTS) {
        float x = pred[3 * i + 0], y = pred[3 * i + 1], z = pred[3 * i + 2];
        predQ[i] = make_float4(-2.f * x, -2.f * y, -2.f * z, 0.f);
        predR[i] = make_float4(x, y, z, x * x + y * y + z * z);
    } else if (i < 2 * NPTS) {
        int j = i - NPTS;
        float x = real[3 * j + 0], y = real[3 * j + 1], z = real[3 * j + 2];
        realQ[j] = make_float4(-2.f * x, -2.f * y, -2.f * z, 0.f);
        realR[j] = make_float4(x, y, z, x * x + y * y + z * z);
    }
}

// ---- main: each wave owns a 16-row strip, sweeps 1024 column tiles ---------
__global__ __launch_bounds__(BLOCK_THREADS) void asd_main(
    const float4* __restrict__ predQ, const float4* __restrict__ predR,
    const float4* __restrict__ realQ, const float4* __restrict__ realR,
    float* __restrict__ partial)
{
    const int lane  = threadIdx.x & 31;
    const int wid   = blockIdx.x * WAVES_PER_BLOCK + (threadIdx.x >> 5);
    const int dir   = wid >> 10;            // 0: pred->real, 1: real->pred
    const int rt    = wid & (NTILES - 1);   // row tile index

    const float4* Q  = dir ? realQ : predQ; // query cloud, Q-form (A matrix)
    const float4* Qr = dir ? realR : predR; // query cloud, R-form (for |q|^2)
    const float4* R  = dir ? predR : realR; // reference cloud, R-form (B matrix)

    const int half = lane >> 4;             // 0: K={0,1}  1: K={2,3}
    const int lm   = lane & 15;

    // A fragment: 16x4 f32, one row per lane (mod 16), K split across halves.
    const float* ap = (const float*)&Q[rt * 16 + lm];
    v2f a;
    a.x = ap[half * 2 + 0];
    a.y = ap[half * 2 + 1];

    // |q|^2 for the 8 rows this (vgpr, half) slot maps to: row = v + 8*half.
    float p2v[8];
#pragma unroll
    for (int v = 0; v < 8; ++v)
        p2v[v] = Qr[rt * 16 + v + half * 8].w;

    float rmin[8];
#pragma unroll
    for (int v = 0; v < 8; ++v) rmin[v] = 3.4e38f;

    // Per-lane base into B data: point (ct*16 + lm), floats (half*2 .. +1).
    const float* rb = (const float*)R + lm * 4 + half * 2;

#pragma unroll 8
    for (int ct = 0; ct < NTILES; ++ct) {
        v2f b = *(const v2f*)(rb + ct * 64);      // 16 pts * 4 floats per tile
        v8f c = {};
        c = __builtin_amdgcn_wmma_f32_16x16x4_f32(
                /*neg_a=*/false, a, /*neg_b=*/false, b,
                /*c_mod=*/(short)0, c, /*reuse_a=*/false, /*reuse_b=*/false);
#pragma unroll
        for (int v = 0; v < 8; ++v)
            rmin[v] = fminf(rmin[v], c[v]);       // running min of raw t
    }

    // Butterfly min within each 16-lane half (columns live on lanes 0..15).
#pragma unroll
    for (int v = 0; v < 8; ++v) {
        float r = rmin[v];
        r = fminf(r, __shfl_xor(r, 1));
        r = fminf(r, __shfl_xor(r, 2));
        r = fminf(r, __shfl_xor(r, 4));
        r = fminf(r, __shfl_xor(r, 8));
        rmin[v] = r;
    }

    // d = sqrt(max(min_t + |q|^2, 0)); lanes 0-15 cover rows 0-7, 16-31 rows 8-15.
    float s = 0.f;
#pragma unroll
    for (int v = 0; v < 8; ++v)
        s += sqrtf(fmaxf(rmin[v] + p2v[v], 0.f));

    s += __shfl_xor(s, 16);                       // combine both halves
    if (lane == 0) partial[wid] = s;
}

// ---- deterministic final reduction -----------------------------------------
__global__ void asd_finalize(const float* __restrict__ partial,
                             float* __restrict__ out)
{
    __shared__ float sm[256];
    int t = threadIdx.x;
    float acc = 0.f;
    for (int i = t; i < TOTAL_WAVES; i += 256) acc += partial[i];
    sm[t] = acc;
    __syncthreads();
    for (int off = 128; off > 0; off >>= 1) {
        if (t < off) sm[t] += sm[t + off];
        __syncthreads();
    }
    if (t == 0) out[0] = sm[0] / (float)(2 * NPTS);
}

// ---------------------------------------------------------------------------
extern "C" void kernel_launch(void* const* d_in, const int* in_sizes, int n_in,
                              void* d_out, int out_size, void* d_ws, size_t ws_size,
                              hipStream_t stream)
{
    const float* real = (const float*)d_in[0];   // setup_inputs order: real_pts
    const float* pred = (const float*)d_in[1];   // then pred_pts

    float4* predQ = (float4*)d_ws;
    float4* predR = predQ + NPTS;
    float4* realQ = predR + NPTS;
    float4* realR = realQ + NPTS;
    float*  partial = (float*)(realR + NPTS);    // TOTAL_WAVES floats

    asd_prep<<<(2 * NPTS + 255) / 256, 256, 0, stream>>>(
        pred, real, predQ, predR, realQ, realR);

    asd_main<<<TOTAL_WAVES / WAVES_PER_BLOCK, BLOCK_THREADS, 0, stream>>>(
        predQ, predR, realQ, realR, partial);

    asd_finalize<<<1, 256, 0, stream>>>(partial, (float*)d_out);
}